// GAT_35115652612106
// MI455X (gfx1250) — compile-verified
//
#include <hip/hip_runtime.h>
#include <math.h>

typedef __attribute__((ext_vector_type(2))) float v2f;
typedef __attribute__((ext_vector_type(8))) float v8f;

#define FDIM 128
#define KCHUNK 32
#define NCHUNK (FDIM / KCHUNK)
#define KSTRIDE 36                  // padded col stride (floats): mult of 4 (b128 LDS align),
                                    // 36*c mod 64 distinct per lane -> conflict-free ds_load_b64
#define CHUNKF (FDIM * KSTRIDE)     // floats per LDS buffer (128 cols x 36)

// ---------------- tiny per-layer transpose: WT[c][k] = W[k][c] ----------------
__global__ void gat_transpose_w(const float* __restrict__ W, float* __restrict__ WT)
{
    int i = blockIdx.x * blockDim.x + threadIdx.x;   // 16384 threads total
    int c = i >> 7;
    int k = i & 127;
    WT[c * FDIM + k] = W[k * FDIM + c];
}

// ---------------- GEMM: h = act(x) @ W via V_WMMA_F32_16X16X4_F32 ----------------
// WT chunks staged global->LDS (column-major image) with CDNA5 async-to-LDS DMA,
// double buffered so the copy of chunk c+1 overlaps the WMMA compute on chunk c.
__global__ __launch_bounds__(256) void gat_gemm_wmma(
    const float* __restrict__ x, const float* __restrict__ WT,
    float* __restrict__ h, int N, int apply_elu)
{
    // Only __shared__ object in this kernel -> LDS offset 0, so the async VDST
    // address is simply the byte index into Wlds.
    __shared__ float Wlds[2][CHUNKF];

    const int tid  = threadIdx.x;
    const int lane = tid & 31;
    const int wave = tid >> 5;
    const int rowBase = blockIdx.x * 128 + wave * 16;

    // Escape handle: passing this pointer into the asm makes Wlds address-taken,
    // so the "memory" clobber legally covers the LDS writes done by the async DMA.
    float* ldsbase = &Wlds[0][0];

    // A-fragment row for this lane (clamped; stores are guarded)
    int arow = rowBase + (lane & 15);
    if (arow >= N) arow = N - 1;
    const int kOffLane = 2 * (lane >> 4);   // lanes 0-15 -> K 0,1 ; lanes 16-31 -> K 2,3
    const int col16 = lane & 15;

    v8f acc[8] = {};

    // async stage of WT[0..127][kc..kc+31] into LDS buffer `buf` (col-major image):
    // 128 cols * 8 float4 along k = 1024 b128 transfers; 256 threads -> 4 each.
    auto stage_async = [&](int kc, int buf) {
        #pragma unroll
        for (int j = 0; j < 4; ++j) {
            int i4  = tid + j * 256;             // [0,1024)
            int col = i4 >> 3;                   // 8 float4 per column
            int q   = i4 & 7;                    // which float4 along k
            unsigned ldsb = (unsigned)((buf * CHUNKF + col * KSTRIDE + q * 4) * 4);
            unsigned goff = (unsigned)((col * FDIM + kc + q * 4) * 4);
            asm volatile("global_load_async_to_lds_b128 %0, %1, %2"
                         :: "v"(ldsb), "v"(goff), "s"(WT), "v"(ldsbase) : "memory");
        }
    };

    stage_async(0, 0);
    asm volatile("s_wait_asynccnt 0x0" ::: "memory");
    __syncthreads();

    for (int c = 0; c < NCHUNK; ++c) {
        if (c + 1 < NCHUNK) stage_async((c + 1) * KCHUNK, (c + 1) & 1);  // overlap copy w/ compute
        const int kc = c * KCHUNK;
        const float* Wb = &Wlds[c & 1][0];

        #pragma unroll 2
        for (int k = 0; k < KCHUNK; k += 4) {
            // A fragment: x[arow][kc+k+kOffLane + {0,1}], ELU fused on load (each elem read once)
            const float* ap = x + (size_t)arow * FDIM + kc + k + kOffLane;
            v2f a;
            a.x = ap[0];
            a.y = ap[1];
            if (apply_elu) {
                a.x = a.x > 0.f ? a.x : (__expf(a.x) - 1.f);
                a.y = a.y > 0.f ? a.y : (__expf(a.y) - 1.f);
            }
            // Batch the 8 independent B-fragment ds_load_b64s so they issue as a
            // cluster (one dscnt wait for the group) instead of load/wait/wmma
            // lock-step on a single reused register pair.
            v2f bf[8];
            #pragma unroll
            for (int nt = 0; nt < 8; ++nt)
                bf[nt] = *(const v2f*)(Wb + (nt * 16 + col16) * KSTRIDE + k + kOffLane);
            #pragma unroll
            for (int nt = 0; nt < 8; ++nt)
                acc[nt] = __builtin_amdgcn_wmma_f32_16x16x4_f32(
                    false, a, false, bf[nt], (short)0, acc[nt], false, false);
        }
        asm volatile("s_wait_asynccnt 0x0" ::: "memory");  // own async copies landed
        __syncthreads();                                   // everyone done reading+copying
    }

    // C layout: VGPR v -> M = v + 8*(lane/16), N = nt*16 + (lane%16)
    const int mOff = 8 * (lane >> 4);
    #pragma unroll
    for (int nt = 0; nt < 8; ++nt) {
        const int col = nt * 16 + col16;
        #pragma unroll
        for (int v = 0; v < 8; ++v) {
            int r = rowBase + mOff + v;
            if (r < N) h[(size_t)r * FDIM + col] = acc[nt][v];
        }
    }
}

// ---------------- per-node attention logits: alpha_src/dst = h . a ----------------
__global__ __launch_bounds__(256) void gat_alpha(
    const float* __restrict__ h, const float* __restrict__ a_src,
    const float* __restrict__ a_dst, float* __restrict__ asrc,
    float* __restrict__ adst, int N)
{
    int node = blockIdx.x * 8 + (threadIdx.x >> 5);
    int lane = threadIdx.x & 31;
    if (node >= N) return;
    const float4 hv = ((const float4*)(h + (size_t)node * FDIM))[lane];
    const float4 sv = ((const float4*)a_src)[lane];
    const float4 dv = ((const float4*)a_dst)[lane];
    float ps = hv.x * sv.x + hv.y * sv.y + hv.z * sv.z + hv.w * sv.w;
    float pd = hv.x * dv.x + hv.y * dv.y + hv.z * dv.z + hv.w * dv.w;
    #pragma unroll
    for (int off = 16; off > 0; off >>= 1) {
        ps += __shfl_xor(ps, off, 32);
        pd += __shfl_xor(pd, off, 32);
    }
    if (lane == 0) { asrc[node] = ps; adst[node] = pd; }
}

// ---------------- init: out = b (broadcast), m = -inf, denom = 0 ----------------
__global__ void gat_init(float* __restrict__ out, const float* __restrict__ b,
                         float* __restrict__ m, float* __restrict__ denom, int N)
{
    int i = blockIdx.x * blockDim.x + threadIdx.x;
    if (i >= N * FDIM) return;
    int f = i & (FDIM - 1);
    out[i] = b[f];
    if (f == 0) { int n = i >> 7; m[n] = -1e30f; denom[n] = 0.f; }
}

// ---------------- edge logits + segment max (float atomic max via int punning) ----------------
__global__ void gat_edge_max(const int* __restrict__ ei, int E, int N,
                             const float* __restrict__ asrc, const float* __restrict__ adst,
                             float* __restrict__ ebuf, float* __restrict__ m)
{
    int e = blockIdx.x * blockDim.x + threadIdx.x;
    int ET = E + N;
    if (e >= ET) return;
    int s, d;
    if (e < E) { s = ei[e]; d = ei[E + e]; } else { s = d = e - E; }
    float val = asrc[s] + adst[d];
    val = val > 0.f ? val : 0.2f * val;           // LeakyReLU(0.2)
    ebuf[e] = val;
    if (val >= 0.f) atomicMax((int*)(m + d), __float_as_int(val));
    else            atomicMin((unsigned int*)(m + d), __float_as_uint(val));
}

// ---------------- exp(e - m[dst]) + segment sum ----------------
__global__ void gat_exp_sum(const int* __restrict__ ei, int E, int N,
                            const float* __restrict__ ebuf, const float* __restrict__ m,
                            float* __restrict__ exbuf, float* __restrict__ denom)
{
    int e = blockIdx.x * blockDim.x + threadIdx.x;
    int ET = E + N;
    if (e >= ET) return;
    int d = (e < E) ? ei[E + e] : e - E;
    float ex = __expf(ebuf[e] - m[d]);
    exbuf[e] = ex;
    unsafeAtomicAdd(denom + d, ex);
}

// ---------------- weighted scatter-aggregate: out[dst] += alpha * h[src] ----------------
__global__ __launch_bounds__(256) void gat_aggregate(
    const int* __restrict__ ei, int E, int N,
    const float* __restrict__ exbuf, const float* __restrict__ denom,
    const float* __restrict__ h, float* __restrict__ out)
{
    int e = blockIdx.x * 8 + (threadIdx.x >> 5);
    int lane = threadIdx.x & 31;
    int ET = E + N;
    if (e >= ET) return;
    int s, d;
    if (e < E) { s = ei[e]; d = ei[E + e]; } else { s = d = e - E; }
    float w = exbuf[e] / denom[d];
    const float4 hv = ((const float4*)(h + (size_t)s * FDIM))[lane];   // L2 hit: h fits in 192MB L2
    float* o = out + (size_t)d * FDIM + lane * 4;
    unsafeAtomicAdd(o + 0, w * hv.x);
    unsafeAtomicAdd(o + 1, w * hv.y);
    unsafeAtomicAdd(o + 2, w * hv.z);
    unsafeAtomicAdd(o + 3, w * hv.w);
}

extern "C" void kernel_launch(void* const* d_in, const int* in_sizes, int n_in,
                              void* d_out, int out_size, void* d_ws, size_t ws_size,
                              hipStream_t stream)
{
    const float* x  = (const float*)d_in[0];
    const int*   ei = (const int*)d_in[1];
    const int N  = in_sizes[0] / FDIM;
    const int E  = in_sizes[1] / 2;
    const int ET = E + N;

    float* ws = (float*)d_ws;
    size_t off = 0;
    float* h     = ws + off; off += (size_t)N * FDIM;
    float* bufA  = ws + off; off += (size_t)N * FDIM;
    float* bufB  = ws + off; off += (size_t)N * FDIM;
    float* WT    = ws + off; off += (size_t)FDIM * FDIM;
    float* asrc  = ws + off; off += N;
    float* adst  = ws + off; off += N;
    float* m     = ws + off; off += N;
    float* den   = ws + off; off += N;
    float* ebuf  = ws + off; off += ET;
    float* exbuf = ws + off; off += ET;

    const float* ins[3]  = { x, bufA, bufB };
    float*       outs[3] = { bufA, bufB, (float*)d_out };

    for (int l = 0; l < 3; ++l) {
        const float* W   = (const float*)d_in[2 + l * 4];
        const float* a_s = (const float*)d_in[3 + l * 4];
        const float* a_d = (const float*)d_in[4 + l * 4];
        const float* b   = (const float*)d_in[5 + l * 4];
        float* out = outs[l];

        int initBlocks = (N * FDIM + 255) / 256;
        gat_init<<<initBlocks, 256, 0, stream>>>(out, b, m, den, N);
        gat_transpose_w<<<(FDIM * FDIM) / 256, 256, 0, stream>>>(W, WT);
        gat_gemm_wmma<<<(N + 127) / 128, 256, 0, stream>>>(ins[l], WT, h, N, l > 0 ? 1 : 0);
        gat_alpha<<<(N + 7) / 8, 256, 0, stream>>>(h, a_s, a_d, asrc, adst, N);
        gat_edge_max<<<(ET + 255) / 256, 256, 0, stream>>>(ei, E, N, asrc, adst, ebuf, m);
        gat_exp_sum<<<(ET + 255) / 256, 256, 0, stream>>>(ei, E, N, ebuf, m, exbuf, den);
        gat_aggregate<<<(ET + 7) / 8, 256, 0, stream>>>(ei, E, N, exbuf, den, h, out);
    }
}